// RegionOfInterestAlignPyramid_3255585210659
// MI455X (gfx1250) — compile-verified
//
#include <hip/hip_runtime.h>
#include <cstdint>

#define EXTENT  7
#define CCH     256
#define THREADS 64

typedef __attribute__((ext_vector_type(4))) float v4f;

// One block = one (box n, output row gy). 64 lanes x float4 = 256 channels.
// All 7 output cells' bilinear source pixels (4 x 1KB each) are issued into
// LDS up-front via gfx1250 async-to-LDS DMA (28 ops, 28KB LDS, ASYNCcnt-
// tracked, in-order retire), then consumed with decremental s_wait_asynccnt.
// Output uses non-temporal b128 stores to keep L2 capacity for the pyramid
// (87MB of feature maps fit in the 192MB L2; the 50MB output is write-once).
__global__ __launch_bounds__(THREADS) void roi_align_pyr_kernel(
    const float* __restrict__ meta,
    const float* __restrict__ boxes,
    const float* __restrict__ p2,
    const float* __restrict__ p3,
    const float* __restrict__ p4,
    const float* __restrict__ p5,
    float* __restrict__ out,
    int N)
{
    __shared__ float lds[EXTENT * 4 * CCH]; // 7 cells x 4 bilinear pixels x 256 ch = 28 KB

    const int tid = threadIdx.x;       // 0..63
    const int blk = blockIdx.x;        // n*7 + gy
    const int n   = blk / EXTENT;
    const int gy  = blk - n * EXTENT;
    if (n >= N) return;

    const float rows = meta[0];
    const float cols = meta[1];

    const float bx1 = boxes[n * 4 + 0];
    const float by1 = boxes[n * 4 + 1];
    const float bx2 = boxes[n * 4 + 2];
    const float by2 = boxes[n * 4 + 3];

    // FPN level selection: clip(4 + round(log2(sqrt(h*w)/sqrt(rows*cols))), 2, 5)
    const float bh = by2 - by1;
    const float bw = bx2 - bx1;
    float rl = log2f(sqrtf(bh * bw) / sqrtf(rows * cols));
    rl = 4.0f + rintf(rl);                 // rintf == round-half-even == jnp.round
    rl = fminf(fmaxf(rl, 2.0f), 5.0f);
    const int li = (int)rl - 2;            // 0..3

    const float* feat; int HW;
    if      (li == 0) { feat = p2; HW = 256; }
    else if (li == 1) { feat = p3; HW = 128; }
    else if (li == 2) { feat = p4; HW = 64;  }
    else              { feat = p5; HW = 32;  }

    const float y1n = by1 / rows, x1n = bx1 / cols;
    const float y2n = by2 / rows, x2n = bx2 / cols;
    const float Hm1 = (float)(HW - 1);

    // Row (y) sample: fixed for this block.
    float ys = (y1n + ((float)gy / 6.0f) * (y2n - y1n)) * Hm1;
    ys = fminf(fmaxf(ys, 0.0f), Hm1);
    const float yfl = floorf(ys);
    const float wy  = ys - yfl;
    const int   y0  = (int)yfl;
    const int   y1i = min(y0 + 1, HW - 1);

    // Column (x) samples for all 7 cells.
    int   x0a[EXTENT], x1a[EXTENT];
    float wxa[EXTENT];
#pragma unroll
    for (int gx = 0; gx < EXTENT; ++gx) {
        float xs = (x1n + ((float)gx / 6.0f) * (x2n - x1n)) * Hm1;
        xs = fminf(fmaxf(xs, 0.0f), Hm1);
        float xfl = floorf(xs);
        wxa[gx] = xs - xfl;
        x0a[gx] = (int)xfl;
        x1a[gx] = min((int)xfl + 1, HW - 1);
    }

    const int cofs = tid * 4; // this lane's channel offset (floats)
    // Generic pointer to LDS object: low 32 bits are the LDS byte offset.
    const uint32_t ldsBase = (uint32_t)(uintptr_t)(&lds[0]) + (uint32_t)(cofs * 4);

    const float* rowA = feat + (size_t)y0  * HW * CCH + cofs;
    const float* rowB = feat + (size_t)y1i * HW * CCH + cofs;

    // Issue ALL cells' bilinear sources (28 async ops per wave; in-order retire).
#pragma unroll
    for (int gx = 0; gx < EXTENT; ++gx) {
        const uint32_t l_ = ldsBase + (uint32_t)(gx * 4 * CCH * 4);
        const float* g00_ = rowA + (size_t)x0a[gx] * CCH;
        const float* g01_ = rowA + (size_t)x1a[gx] * CCH;
        const float* g10_ = rowB + (size_t)x0a[gx] * CCH;
        const float* g11_ = rowB + (size_t)x1a[gx] * CCH;
        asm volatile("global_load_async_to_lds_b128 %0, %1, off" :: "v"(l_),         "v"(g00_) : "memory");
        asm volatile("global_load_async_to_lds_b128 %0, %1, off" :: "v"(l_ + 1024u), "v"(g01_) : "memory");
        asm volatile("global_load_async_to_lds_b128 %0, %1, off" :: "v"(l_ + 2048u), "v"(g10_) : "memory");
        asm volatile("global_load_async_to_lds_b128 %0, %1, off" :: "v"(l_ + 3072u), "v"(g11_) : "memory");
    }

    float* outBase = out + ((size_t)n * EXTENT + gy) * EXTENT * CCH + cofs;
    const float omwy = 1.0f - wy;

    // Consume one cell per step; async loads retire in order, so waiting for
    // ASYNCcnt <= 4*(6-gx) guarantees cell gx's 4 pixels are resident in LDS.
#define CELL(gxi, cnt_lit) do {                                                      \
        asm volatile("s_wait_asynccnt " #cnt_lit ::: "memory");                      \
        const float* lp = &lds[(gxi) * 4 * CCH + cofs];                              \
        const v4f f00 = *reinterpret_cast<const v4f*>(lp);                           \
        const v4f f01 = *reinterpret_cast<const v4f*>(lp + CCH);                     \
        const v4f f10 = *reinterpret_cast<const v4f*>(lp + 2 * CCH);                 \
        const v4f f11 = *reinterpret_cast<const v4f*>(lp + 3 * CCH);                 \
        const float wx = wxa[gxi], omwx = 1.0f - wx;                                 \
        v4f r;                                                                       \
        r.x = (f00.x * omwx + f01.x * wx) * omwy + (f10.x * omwx + f11.x * wx) * wy; \
        r.y = (f00.y * omwx + f01.y * wx) * omwy + (f10.y * omwx + f11.y * wx) * wy; \
        r.z = (f00.z * omwx + f01.z * wx) * omwy + (f10.z * omwx + f11.z * wx) * wy; \
        r.w = (f00.w * omwx + f01.w * wx) * omwy + (f10.w * omwx + f11.w * wx) * wy; \
        __builtin_nontemporal_store(r, reinterpret_cast<v4f*>(outBase + (gxi) * CCH)); \
    } while (0)

    CELL(0, 24);
    CELL(1, 20);
    CELL(2, 16);
    CELL(3, 12);
    CELL(4, 8);
    CELL(5, 4);
    CELL(6, 0);
#undef CELL
}

extern "C" void kernel_launch(void* const* d_in, const int* in_sizes, int n_in,
                              void* d_out, int out_size, void* d_ws, size_t ws_size,
                              hipStream_t stream) {
    (void)n_in; (void)out_size; (void)d_ws; (void)ws_size;
    const float* meta  = (const float*)d_in[0];
    const float* boxes = (const float*)d_in[1];
    const float* p2    = (const float*)d_in[2];
    const float* p3    = (const float*)d_in[3];
    const float* p4    = (const float*)d_in[4];
    const float* p5    = (const float*)d_in[5];
    float* out = (float*)d_out;

    const int N = in_sizes[1] / 4; // boxes: [1, N, 4]
    dim3 grid(N * EXTENT);
    dim3 block(THREADS);
    hipLaunchKernelGGL(roi_align_pyr_kernel, grid, block, 0, stream,
                       meta, boxes, p2, p3, p4, p5, out, N);
}